// CLIPVisionTower_VisionZip_17437567222419
// MI455X (gfx1250) — compile-verified
//
#include <hip/hip_runtime.h>

typedef __attribute__((ext_vector_type(2))) float v2f;
typedef __attribute__((ext_vector_type(8))) float v8f;

#define B_    32
#define H_    16
#define N_    577
#define D_    1024
#define DM_   64
#define DOM_  54
#define NSEL  55      // 1 cls + 54 dominant
#define CTX_  10
#define CTXP  16      // padded cluster count (zero rows 10..15)
#define REM_  522     // 577 - 55
#define STEP_ 52      // max(1, 522 // 10)
#define MRG_  512     // 522 - 10
#define NOUT  65      // 55 + 10

// ---------------------------------------------------------------------------
// K1: per-batch head-sum of attention row 0, iterative top-54 (ties -> lower
// index, matching jax.lax.top_k), build sel/dom/torig/morig index lists.
// ---------------------------------------------------------------------------
__global__ __launch_bounds__(256) void k1_select(const float* __restrict__ attn,
                                                 int* __restrict__ sel,
                                                 int* __restrict__ dom,
                                                 int* __restrict__ torig,
                                                 int* __restrict__ morig) {
    const int b = blockIdx.x;
    const int tid = threadIdx.x;

    __shared__ float vals[576];
    __shared__ float rv[256];
    __shared__ int   ri[256];
    __shared__ int   ssel[NSEL];
    __shared__ unsigned char flag[N_];

    // cls_attention_sum[j] = sum_h attn[b,h,0,1+j]
    for (int j = tid; j < 576; j += 256) {
        float s = 0.f;
        const size_t base = ((size_t)b * H_) * (size_t)N_ * (size_t)N_;
        for (int h = 0; h < H_; ++h)
            s += attn[base + (size_t)h * N_ * N_ + 1 + j];
        vals[j] = s;
    }
    for (int j = tid; j < N_; j += 256) flag[j] = 0;
    if (tid == 0) ssel[0] = 0;
    __syncthreads();

    // 54 rounds of block-wide argmax (value desc, index asc tiebreak)
    for (int round = 0; round < DOM_; ++round) {
        float bv = -3.402823466e+38f;
        int   bi = 0x7fffffff;
        for (int j = tid; j < 576; j += 256) {
            float v = vals[j];
            if (v > bv || (v == bv && j < bi)) { bv = v; bi = j; }
        }
        rv[tid] = bv; ri[tid] = bi;
        __syncthreads();
        for (int s = 128; s > 0; s >>= 1) {
            if (tid < s) {
                float ov = rv[tid + s]; int oi = ri[tid + s];
                if (ov > rv[tid] || (ov == rv[tid] && oi < ri[tid])) {
                    rv[tid] = ov; ri[tid] = oi;
                }
            }
            __syncthreads();
        }
        if (tid == 0) {
            int j = ri[0];
            ssel[1 + round] = j + 1;      // +1: skip cls position
            vals[j] = -3.402823466e+38f;  // mask out
        }
        __syncthreads();
    }

    // all_indices (unsorted, as reference returns) + flags
    if (tid < NSEL) {
        sel[b * NSEL + tid] = ssel[tid];
        flag[ssel[tid]] = 1;
    }
    __syncthreads();

    // sorted dominant indices via rank (indices are distinct)
    if (tid < NSEL) {
        int v = ssel[tid];
        int rank = 0;
        for (int j = 0; j < NSEL; ++j) rank += (ssel[j] < v) ? 1 : 0;
        dom[b * NSEL + rank] = v;
    }

    // remaining tokens in increasing order; split into targets / merge list
    if (tid == 0) {
        int p = 0, m = 0;
        for (int j = 0; j < N_; ++j) {
            if (!flag[j]) {
                if ((p % STEP_) == 0 && (p / STEP_) < CTX_)
                    torig[b * CTX_ + p / STEP_] = j;
                else
                    morig[b * MRG_ + (m++)] = j;
                ++p;
            }
        }
    }
}

// ---------------------------------------------------------------------------
// K2: normalize the 10 target metric vectors (64-dim); zero-fill the padding
// rows 10..15 so K3's B-operand loads need no bounds guard.
// ---------------------------------------------------------------------------
__global__ __launch_bounds__(64) void k2_norm(const float* __restrict__ metric,
                                              const int* __restrict__ torig,
                                              float* __restrict__ tnorm) {
    const int c = blockIdx.x, b = blockIdx.y, t = threadIdx.x;
    float* dst = tnorm + ((size_t)b * CTXP + c) * DM_ + t;
    if (c >= CTX_) { *dst = 0.f; return; }
    const int tok = torig[b * CTX_ + c];
    const float v = metric[((size_t)b * N_ + tok) * DM_ + t];
    __shared__ float red[64];
    red[t] = v * v;
    __syncthreads();
    for (int s = 32; s > 0; s >>= 1) {
        if (t < s) red[t] += red[t + s];
        __syncthreads();
    }
    const float nrm = sqrtf(red[0]);
    *dst = v / nrm;
}

// ---------------------------------------------------------------------------
// K3: similarity (512x64)x(64x10pad16) per batch via V_WMMA_F32_16X16X4_F32,
// then per-row argmax over the 10 clusters. One wave32 per 16-row tile.
// Row scaling of merge vectors does not change argmax, so A is raw metric.
// ---------------------------------------------------------------------------
__global__ __launch_bounds__(32) void k3_sim(const float* __restrict__ metric,
                                             const int* __restrict__ morig,
                                             const float* __restrict__ tnorm,
                                             int* __restrict__ assign) {
    const int tile = blockIdx.x;           // 0..31 (M tiles)
    const int b    = blockIdx.y;
    const int lane = threadIdx.x;
    const int mrow  = lane & 15;
    const int khalf = (lane < 16) ? 0 : 2;
    const int n     = lane & 15;
    const int m_base = tile * 16;

    const int tokA = morig[b * MRG_ + m_base + mrow];
    const float* mA = metric + ((size_t)b * N_ + tokA) * DM_;
    const float* tB = tnorm + (size_t)b * CTXP * DM_ + (size_t)n * DM_;

    v8f acc = {};
#pragma unroll
    for (int k0 = 0; k0 < DM_; k0 += 4) {
        v2f a, bm;
        a.x  = mA[k0 + khalf];
        a.y  = mA[k0 + khalf + 1];
        bm.x = tB[k0 + khalf];
        bm.y = tB[k0 + khalf + 1];
        acc = __builtin_amdgcn_wmma_f32_16x16x4_f32(
            false, a, false, bm, (short)0, acc, false, false);
    }

    // scatter D tile (rows M=0..15, cols N=0..15) to LDS, argmax per row
    __shared__ float sim[16 * 16];
    union { v8f v; float f[8]; } u;
    u.v = acc;
#pragma unroll
    for (int r = 0; r < 8; ++r) {
        const int row = (lane < 16) ? r : (r + 8);
        sim[row * 16 + (lane & 15)] = u.f[r];
    }
    __syncthreads();
    if (lane < 16) {
        float best = sim[lane * 16];
        int bc = 0;
        for (int c = 1; c < CTX_; ++c) {
            float v = sim[lane * 16 + c];
            if (v > best) { best = v; bc = c; }   // first-max tiebreak
        }
        assign[b * MRG_ + m_base + lane] = bc;
    }
}

// ---------------------------------------------------------------------------
// K4: cluster counts, clamped to >= 1.
// ---------------------------------------------------------------------------
__global__ __launch_bounds__(256) void k4_counts(const int* __restrict__ assign,
                                                 float* __restrict__ counts) {
    const int b = blockIdx.x, t = threadIdx.x;
    __shared__ int cnt[CTX_];
    if (t < CTX_) cnt[t] = 0;
    __syncthreads();
    for (int m = t; m < MRG_; m += 256)
        atomicAdd(&cnt[assign[b * MRG_ + m]], 1);
    __syncthreads();
    if (t < CTX_) counts[b * CTX_ + t] = fmaxf(1.0f, (float)cnt[t]);
}

// ---------------------------------------------------------------------------
// K5: aggregated = one_hot^T(10x512) x hidden_merge(512x1024) via
// V_WMMA_F32_16X16X4_F32; A generated on the fly as exact 0/1 from the
// assignment (staged in LDS). One wave32 per (batch, 16-col D tile).
// Divides by counts, adds target hidden row, writes contextual rows 55..64.
// ---------------------------------------------------------------------------
__global__ __launch_bounds__(32) void k5_agg(const float* __restrict__ hidden,
                                             const int* __restrict__ morig,
                                             const int* __restrict__ torig,
                                             const int* __restrict__ assign,
                                             const float* __restrict__ counts,
                                             float* __restrict__ out) {
    const int ntile = blockIdx.x;           // 0..63 (D tiles)
    const int b     = blockIdx.y;
    const int lane  = threadIdx.x;
    const int khalf = (lane < 16) ? 0 : 2;
    const int n     = lane & 15;
    const int dbase = ntile * 16;
    const int myc   = lane & 15;            // cluster row (M) this lane feeds

    __shared__ int s_as[MRG_];
    __shared__ int s_mo[MRG_];
    {
        const int* as = assign + b * MRG_;
        const int* mo = morig + b * MRG_;
        for (int i = lane; i < MRG_; i += 32) {
            s_as[i] = as[i];
            s_mo[i] = mo[i];
        }
    }
    __syncthreads();

    const float* hb = hidden + (size_t)b * N_ * D_;

    v8f acc = {};
#pragma unroll 4
    for (int kb = 0; kb < MRG_; kb += 4) {
        v2f a, bm;
        const int a0 = s_as[kb + khalf];
        const int a1 = s_as[kb + khalf + 1];
        a.x = (a0 == myc) ? 1.f : 0.f;
        a.y = (a1 == myc) ? 1.f : 0.f;
        const int t0 = s_mo[kb + khalf];
        const int t1 = s_mo[kb + khalf + 1];
        bm.x = hb[(size_t)t0 * D_ + dbase + n];
        bm.y = hb[(size_t)t1 * D_ + dbase + n];
        acc = __builtin_amdgcn_wmma_f32_16x16x4_f32(
            false, a, false, bm, (short)0, acc, false, false);
    }

    union { v8f v; float f[8]; } u;
    u.v = acc;
#pragma unroll
    for (int r = 0; r < 8; ++r) {
        const int c = (lane < 16) ? r : (r + 8);
        if (c < CTX_) {
            const float agg = u.f[r] / counts[b * CTX_ + c];
            const int tok = torig[b * CTX_ + c];
            const float hid = hb[(size_t)tok * D_ + dbase + n];
            out[((size_t)b * NOUT + NSEL + c) * D_ + dbase + n] = agg + hid;
        }
    }
}

// ---------------------------------------------------------------------------
// K6: dominant-token gather copy (sorted indices, float4 = B128 path)
// + all_indices output (written as float after the hidden block).
// ---------------------------------------------------------------------------
__global__ __launch_bounds__(256) void k6_dom(const float* __restrict__ hidden,
                                              const int* __restrict__ dom,
                                              const int* __restrict__ sel,
                                              float* __restrict__ out) {
    const int j = blockIdx.x, b = blockIdx.y, t = threadIdx.x;
    const int tok = dom[b * NSEL + j];
    const float4* src = (const float4*)(hidden + ((size_t)b * N_ + tok) * D_);
    float4* dst = (float4*)(out + ((size_t)b * NOUT + j) * D_);
    // D_/4 = 256 float4 per row; one per thread
    dst[t] = src[t];
    if (j == 0 && t < NSEL)
        out[(size_t)B_ * NOUT * D_ + b * NSEL + t] = (float)sel[b * NSEL + t];
}

// ---------------------------------------------------------------------------
extern "C" void kernel_launch(void* const* d_in, const int* in_sizes, int n_in,
                              void* d_out, int out_size, void* d_ws, size_t ws_size,
                              hipStream_t stream) {
    const float* attn   = (const float*)d_in[0];
    const float* hidden = (const float*)d_in[1];
    const float* metric = (const float*)d_in[2];
    float* out = (float*)d_out;

    char* ws = (char*)d_ws;
    int*   sel    = (int*)ws;   ws += (size_t)B_ * NSEL * 4;
    int*   dom    = (int*)ws;   ws += (size_t)B_ * NSEL * 4;
    int*   torig  = (int*)ws;   ws += (size_t)B_ * CTX_ * 4;
    int*   morig  = (int*)ws;   ws += (size_t)B_ * MRG_ * 4;
    int*   assign = (int*)ws;   ws += (size_t)B_ * MRG_ * 4;
    float* counts = (float*)ws; ws += (size_t)B_ * CTX_ * 4;
    float* tnorm  = (float*)ws; ws += (size_t)B_ * CTXP * DM_ * 4;

    k1_select<<<B_, 256, 0, stream>>>(attn, sel, dom, torig, morig);
    k2_norm<<<dim3(CTXP, B_), 64, 0, stream>>>(metric, torig, tnorm);
    k3_sim<<<dim3(MRG_ / 16, B_), 32, 0, stream>>>(metric, morig, tnorm, assign);
    k4_counts<<<B_, 256, 0, stream>>>(assign, counts);
    k5_agg<<<dim3(D_ / 16, B_), 32, 0, stream>>>(hidden, morig, torig, assign, counts, out);
    k6_dom<<<dim3(NSEL, B_), 256, 0, stream>>>(hidden, dom, sel, out);
}